// LocalGrouper_21397527069088
// MI455X (gfx1250) — compile-verified
//
#include <hip/hip_runtime.h>
#include <math.h>

// Problem constants (reference: b=16, n=4096, s=1024, d=128, k=24)
#define B_   16
#define N_   4096
#define S_   1024
#define D_   128
#define K_   24

// d_out flat layout (floats), in reference return order
static constexpr size_t OFF0 = 0;                              // xyz_sampled (b,s,3)
static constexpr size_t OFF1 = (size_t)B_ * S_ * 3;            // feat_sampled (b,s,128)
static constexpr size_t OFF2 = OFF1 + (size_t)B_ * S_ * D_;    // xyz_knn (b,s,24,3)
static constexpr size_t OFF3 = OFF2 + (size_t)B_ * S_ * K_ * 3;// feat_knn (b,s,24,256)

typedef __attribute__((ext_vector_type(2))) float v2f;
typedef __attribute__((ext_vector_type(8))) float v8f;

// ---------------------------------------------------------------- K0: init
__global__ void init_stats(double* stats) {
  if (threadIdx.x < 4) stats[threadIdx.x] = 0.0;
}

// ---------------------------------------------------------------- K1: FPS
// One block per batch; 1024 threads (32 waves), 4 points/thread in registers.
__global__ __launch_bounds__(1024) void fps_kernel(const float* __restrict__ xyz,
                                                   int* __restrict__ fps_idx) {
  const int b = blockIdx.x;
  const int tid = threadIdx.x;
  const int lane = tid & 31, wid = tid >> 5;
  const float* xb = xyz + (size_t)b * N_ * 3;

  float px[4], py[4], pz[4], dd[4];
#pragma unroll
  for (int j = 0; j < 4; ++j) {
    int p = tid * 4 + j;
    px[j] = xb[p * 3 + 0]; py[j] = xb[p * 3 + 1]; pz[j] = xb[p * 3 + 2];
    dd[j] = 1e10f;
  }

  __shared__ int s_cur;
  __shared__ float s_v[32];
  __shared__ int s_i[32];
  if (tid == 0) s_cur = 0;
  __syncthreads();

  for (int it = 0; it < S_; ++it) {
    int cur = s_cur;
    if (tid == 0) fps_idx[b * S_ + it] = cur;
    float cx = xb[cur * 3 + 0], cy = xb[cur * 3 + 1], cz = xb[cur * 3 + 2];
    float mv = -1.0f; int mi = 0;
#pragma unroll
    for (int j = 0; j < 4; ++j) {
      float dx = px[j] - cx, dy = py[j] - cy, dz = pz[j] - cz;
      float d = dx * dx + dy * dy + dz * dz;
      dd[j] = fminf(dd[j], d);
      if (dd[j] > mv) { mv = dd[j]; mi = tid * 4 + j; }   // strict > keeps lowest idx
    }
#pragma unroll
    for (int off = 16; off > 0; off >>= 1) {              // wave32 butterfly argmax
      float ov = __shfl_xor(mv, off, 32);
      int   oi = __shfl_xor(mi, off, 32);
      if (ov > mv || (ov == mv && oi < mi)) { mv = ov; mi = oi; }
    }
    if (lane == 0) { s_v[wid] = mv; s_i[wid] = mi; }
    __syncthreads();
    if (wid == 0) {
      mv = s_v[lane]; mi = s_i[lane];
#pragma unroll
      for (int off = 16; off > 0; off >>= 1) {
        float ov = __shfl_xor(mv, off, 32);
        int   oi = __shfl_xor(mi, off, 32);
        if (ov > mv || (ov == mv && oi < mi)) { mv = ov; mi = oi; }
      }
      if (lane == 0) s_cur = mi;
    }
    __syncthreads();
  }
}

// ---------------------------------------------------------------- K2: gather sampled
__global__ __launch_bounds__(128) void gather_sampled(const float* __restrict__ xyz,
                                                      const float* __restrict__ feat,
                                                      const int* __restrict__ fps_idx,
                                                      float* __restrict__ out) {
  const int row = blockIdx.x;       // b*S + s
  const int b = row >> 10;
  const int tid = threadIdx.x;
  const int id = fps_idx[row];
  float* xyz_s  = out + OFF0;
  float* feat_s = out + OFF1;
  feat_s[(size_t)row * D_ + tid] = feat[((size_t)b * N_ + id) * D_ + tid];
  if (tid < 3) xyz_s[row * 3 + tid] = xyz[((size_t)b * N_ + id) * 3 + tid];
}

// ---------------------------------------------------------------- K3: kNN via WMMA
// Block = 16 waves = 512 threads; handles 16 queries vs all 4096 points.
// Phase 1: V_WMMA_F32_16X16X4_F32 computes 16x16 dot tiles (K=3 padded to 4);
//          full distance rows land in a 16 x (4096+8) f32 dynamic-LDS buffer.
// Phase 2: each wave does 24 iterative argmins over its query row (ties->lowest idx).
#define KNN_RSTRIDE (N_ + 8)
__global__ __launch_bounds__(512) void knn_kernel(const float* __restrict__ xyz,
                                                  const float* __restrict__ out,
                                                  int* __restrict__ idx_knn) {
  extern __shared__ float sdist[];      // 16 * KNN_RSTRIDE floats (~263 KB LDS)
  __shared__ float qn_s[16];
  const int bq = blockIdx.x;
  const int b = bq >> 6;                // 64 query tiles per batch
  const int qbase = (bq & 63) << 4;
  const int tid = threadIdx.x;
  const int lane = tid & 31;
  const int w = tid >> 5;
  const float* xb = xyz + (size_t)b * N_ * 3;
  const float* xyz_s = out + OFF0;

  // A matrix: 16 queries x K(=4, last zero). Lanes 0-15 -> {K0,K1}, 16-31 -> {K2,0}
  const int qm = qbase + (lane & 15);
  const float qx = xyz_s[((b << 10) + qm) * 3 + 0];
  const float qy = xyz_s[((b << 10) + qm) * 3 + 1];
  const float qz = xyz_s[((b << 10) + qm) * 3 + 2];
  if (tid < 16) qn_s[tid] = qx * qx + qy * qy + qz * qz;
  v2f Am;
  if (lane < 16) { Am.x = qx; Am.y = qy; } else { Am.x = qz; Am.y = 0.0f; }
  __syncthreads();

  const int mhi = (lane >= 16) ? 8 : 0;
  // Scalarize the wave-uniform loop bound so EXEC stays all-ones around WMMA.
  const int wu = __builtin_amdgcn_readfirstlane(w);
  for (int t = wu; t < N_ / 16; t += 16) {      // 16 point-tiles per wave
    const int pbase = t << 4;
    const int p = pbase + (lane & 15);
    const float x = xb[p * 3 + 0], y = xb[p * 3 + 1], z = xb[p * 3 + 2];
    const float xn = x * x + y * y + z * z;
    v2f Bm;                                     // B: K x 16 points, same lane split
    if (lane < 16) { Bm.x = x; Bm.y = y; } else { Bm.x = z; Bm.y = 0.0f; }
    v8f c = {};
    c = __builtin_amdgcn_wmma_f32_16x16x4_f32(false, Am, false, Bm,
                                              (short)0, c, false, false);
#pragma unroll
    for (int r = 0; r < 8; ++r) {               // d = (|q|^2 - 2*dot) + |x|^2
      float d = fmaf(-2.0f, c[r], qn_s[r + mhi]) + xn;
      sdist[(r + mhi) * KNN_RSTRIDE + pbase + (lane & 15)] = d;
    }
  }
  __syncthreads();

  float* row = sdist + w * KNN_RSTRIDE;
  int* out_idx = idx_knn + (size_t)((b << 10) + qbase + w) * K_;
  for (int iter = 0; iter < K_; ++iter) {
    float mv = __builtin_inff(); int mi = 0x7fffffff;
#pragma unroll 4
    for (int j = 0; j < N_ / 32; ++j) {
      int p = (j << 5) + lane;
      float v = row[p];
      if (v < mv || (v == mv && p < mi)) { mv = v; mi = p; }
    }
#pragma unroll
    for (int off = 16; off > 0; off >>= 1) {
      float ov = __shfl_xor(mv, off, 32);
      int   oi = __shfl_xor(mi, off, 32);
      if (ov < mv || (ov == mv && oi < mi)) { mv = ov; mi = oi; }
    }
    if (lane == 0) out_idx[iter] = mi;
    if (lane == (mi & 31)) row[mi] = __builtin_inff();  // remove winner
  }
}

// ---------------------------------------------------------------- K4: stats pass
// Gather-only pass: feat (32 MB) is L2-resident, so this costs ~no HBM traffic.
// Accumulates f64 sum/sumsq of the centered values (feat and xyz) via atomics.
__global__ __launch_bounds__(128) void stats_pass(const float* __restrict__ xyz,
                                                  const float* __restrict__ feat,
                                                  const int* __restrict__ idx_knn,
                                                  const float* __restrict__ out,
                                                  double* __restrict__ stats) {
  const int row = blockIdx.x;       // b*S + s
  const int b = row >> 10;
  const int tid = threadIdx.x;
  const float* xyz_s  = out + OFF0;
  const float* feat_s = out + OFF1;
  const float* xb = xyz + (size_t)b * N_ * 3;

  __shared__ float xs[4];
  __shared__ int nidx[K_];
  const float fsv = feat_s[(size_t)row * D_ + tid];
  if (tid < 3) xs[tid] = xyz_s[row * 3 + tid];
  if (tid < K_) nidx[tid] = idx_knn[(size_t)row * K_ + tid];
  __syncthreads();

  double fsum = 0.0, fsq = 0.0, xsum = 0.0, xsq = 0.0;
  for (int k = 0; k < K_; ++k) {
    const int id = nidx[k];
    const float c = feat[((size_t)b * N_ + id) * D_ + tid] - fsv;
    fsum += c; fsq += (double)c * c;
    if (tid < 3) {
      const float xc = xb[(size_t)id * 3 + tid] - xs[tid];
      xsum += xc; xsq += (double)xc * xc;
    }
  }

  __shared__ double rbuf[128];
  double vals[4] = {fsum, fsq, xsum, xsq};
  double acc[4];
  for (int v = 0; v < 4; ++v) {
    rbuf[tid] = vals[v];
    __syncthreads();
    for (int s = 64; s > 0; s >>= 1) {
      if (tid < s) rbuf[tid] += rbuf[tid + s];
      __syncthreads();
    }
    if (tid == 0) acc[v] = rbuf[0];
    __syncthreads();
  }
  if (tid == 0) {
    atomicAdd(&stats[0], acc[0]);   // global_atomic_add_f64
    atomicAdd(&stats[1], acc[1]);
    atomicAdd(&stats[2], acc[2]);
    atomicAdd(&stats[3], acc[3]);
  }
}

// ---------------------------------------------------------------- K5: denominators
__global__ void finalize_scales(const double* __restrict__ stats,
                                float* __restrict__ denoms) {
  if (threadIdx.x == 0) {
    const double fN = (double)B_ * S_ * K_ * D_;
    const double xN = (double)B_ * S_ * K_ * 3;
    double fvar = (stats[1] - stats[0] * stats[0] / fN) / (fN - 1.0);
    double xvar = (stats[3] - stats[2] * stats[2] / xN) / (xN - 1.0);
    double fstd = sqrt(fvar > 0.0 ? fvar : 0.0);
    double xstd = sqrt(xvar > 0.0 ? xvar : 0.0);
    denoms[0] = (float)fstd + 1e-5f;   // match reference: fc / (std + eps)
    denoms[1] = (float)xstd + 1e-5f;
  }
}

// ---------------------------------------------------------------- K6: write pass
// Second gather (L2 hits), normalize with IEEE f32 division (matches reference
// rounding), write final outputs once with non-temporal stores (write-once data;
// keep feat resident in L2).
__global__ __launch_bounds__(128) void write_pass(const float* __restrict__ xyz,
                                                  const float* __restrict__ feat,
                                                  const int* __restrict__ idx_knn,
                                                  const float* __restrict__ denoms,
                                                  float* __restrict__ out) {
  const int row = blockIdx.x;       // b*S + s
  const int b = row >> 10;
  const int tid = threadIdx.x;
  const float* xyz_s  = out + OFF0;
  const float* feat_s = out + OFF1;
  float* out_xknn = out + OFF2;
  float* out_fknn = out + OFF3;
  const float* xb = xyz + (size_t)b * N_ * 3;

  __shared__ float xs[4];
  __shared__ int nidx[K_];
  const float fsv = feat_s[(size_t)row * D_ + tid];
  if (tid < 3) xs[tid] = xyz_s[row * 3 + tid];
  if (tid < K_) nidx[tid] = idx_knn[(size_t)row * K_ + tid];
  __syncthreads();

  const float fden = denoms[0];
  const float xden = denoms[1];
  for (int k = 0; k < K_; ++k) {
    const int id = nidx[k];
    const float c = feat[((size_t)b * N_ + id) * D_ + tid] - fsv;
    const size_t base = ((size_t)row * K_ + k) * (2 * D_);
    __builtin_nontemporal_store(c / fden, &out_fknn[base + tid]);
    __builtin_nontemporal_store(fsv,      &out_fknn[base + D_ + tid]);
    if (tid < 3) {
      const float xc = xb[(size_t)id * 3 + tid] - xs[tid];
      __builtin_nontemporal_store(xc / xden,
                                  &out_xknn[((size_t)row * K_ + k) * 3 + tid]);
    }
  }
}

// ---------------------------------------------------------------- launch
extern "C" void kernel_launch(void* const* d_in, const int* in_sizes, int n_in,
                              void* d_out, int out_size, void* d_ws, size_t ws_size,
                              hipStream_t stream) {
  const float* xyz  = (const float*)d_in[0];   // (16,4096,3) f32
  const float* feat = (const float*)d_in[1];   // (16,4096,128) f32
  float* out = (float*)d_out;

  char* ws = (char*)d_ws;
  int*    fps_idx = (int*)ws;                               // 65,536 B
  int*    idx_knn = (int*)(ws + 65536);                     // 1,572,864 B
  double* stats   = (double*)(ws + 65536 + 1572864);        // 32 B (8B aligned)
  float*  denoms  = (float*)(ws + 65536 + 1572864 + 32);    // 8 B

  init_stats<<<1, 4, 0, stream>>>(stats);
  fps_kernel<<<B_, 1024, 0, stream>>>(xyz, fps_idx);
  gather_sampled<<<B_ * S_, D_, 0, stream>>>(xyz, feat, fps_idx, out);
  const size_t shbytes = (size_t)16 * KNN_RSTRIDE * sizeof(float);
  knn_kernel<<<B_ * (S_ / 16), 512, shbytes, stream>>>(xyz, out, idx_knn);
  stats_pass<<<B_ * S_, D_, 0, stream>>>(xyz, feat, idx_knn, out, stats);
  finalize_scales<<<1, 32, 0, stream>>>(stats, denoms);
  write_pass<<<B_ * S_, D_, 0, stream>>>(xyz, feat, idx_knn, denoms, out);
}